// NodeClassifierConv_66030827209235
// MI455X (gfx1250) — compile-verified
//
#include <hip/hip_runtime.h>
#include <hip/hip_bf16.h>

typedef __attribute__((ext_vector_type(2))) float v2f;
typedef __attribute__((ext_vector_type(8))) float v8f;

// ---------------------------------------------------------------------------
// Degree histogram: deg[idx[e]] += 1  (float, atomics)
// ---------------------------------------------------------------------------
__global__ void deg_count_kernel(const int* __restrict__ idx,
                                 float* __restrict__ deg, int n) {
  int i = blockIdx.x * blockDim.x + threadIdx.x;
  if (i < n) atomicAdd(&deg[idx[i]], 1.0f);
}

// ---------------------------------------------------------------------------
// Y[N x 128] = (X * rsqrt(max(deg,1))[:,None]) @ W[128 x 128]
// One wave32 computes one 16x16 output tile via V_WMMA_F32_16X16X4_F32,
// chaining K=128 as 32 MAC steps. Block = 256 threads = 8 waves = the 8
// column tiles of the 128-wide output. grid.x = N/16 (N % 16 == 0 here).
// EXEC stays all-ones (WMMA requirement): no divergent guards.
// ---------------------------------------------------------------------------
__global__ __launch_bounds__(256) void gemm_norm_wmma_kernel(
    const float* __restrict__ X, const float* __restrict__ deg,
    const float* __restrict__ W, float* __restrict__ Y) {
  const int lane = threadIdx.x & 31;
  const int wave = threadIdx.x >> 5;   // column tile 0..7
  const int half = lane >> 4;          // 0 or 1
  const int l15  = lane & 15;
  const int row  = blockIdx.x * 16 + l15;   // A row owned by this lane
  const int col  = (wave << 4) + l15;       // B/D column owned by this lane

  const float rs = rsqrtf(fmaxf(deg[row], 1.0f));
  const v2f* __restrict__ A2 = (const v2f*)(X + (size_t)row * 128);

  v8f c = {};
#pragma unroll
  for (int kk = 0; kk < 32; ++kk) {
    const int k = (kk << 2) + (half << 1);        // K base for this lane
    v2f a = A2[(kk << 1) + half];                 // X[row][k], X[row][k+1]
    a.x *= rs;
    a.y *= rs;
    v2f b;
    b.x = W[(size_t)k * 128 + col];               // W[k][col]
    b.y = W[(size_t)(k + 1) * 128 + col];         // W[k+1][col]
    // D = A(16x4,f32) * B(4x16,f32) + C(16x16,f32)
    c = __builtin_amdgcn_wmma_f32_16x16x4_f32(
        /*neg_a=*/false, a, /*neg_b=*/false, b,
        /*c_mod=*/(short)0, c, /*reuse_a=*/false, /*reuse_b=*/false);
  }

  float* __restrict__ Yt = Y + (size_t)blockIdx.x * 16 * 128;
#pragma unroll
  for (int i = 0; i < 8; ++i) {
    // C/D layout: VGPR i -> M = i (lanes 0-15) / M = i+8 (lanes 16-31)
    Yt[(size_t)(i + half * 8) * 128 + col] = c[i];
  }
}

// ---------------------------------------------------------------------------
// Edge scatter:  agg[dst[e]] += feat[src[e]] * rsqrt(max(indeg[dst[e]],1))
// One wave32 per edge; each lane moves a float4 (128 floats per edge row),
// accumulating with global_atomic_add_f32.
// ---------------------------------------------------------------------------
__global__ __launch_bounds__(256) void scatter_edges_kernel(
    const float* __restrict__ feat, const int* __restrict__ src,
    const int* __restrict__ dst, const float* __restrict__ indeg,
    float* __restrict__ agg, int nE) {
  const int e    = (blockIdx.x * blockDim.x + threadIdx.x) >> 5;
  const int lane = threadIdx.x & 31;
  if (e >= nE) return;
  const int   s  = src[e];
  const int   d  = dst[e];
  const float sc = rsqrtf(fmaxf(indeg[d], 1.0f));
  const float4 v = ((const float4*)(feat + (size_t)s * 128))[lane];
  float* o = agg + (size_t)d * 128 + lane * 4;
  atomicAdd(o + 0, v.x * sc);
  atomicAdd(o + 1, v.y * sc);
  atomicAdd(o + 2, v.z * sc);
  atomicAdd(o + 3, v.w * sc);
}

// ---------------------------------------------------------------------------
// In-place bias add (+ optional second bias) and optional leaky ReLU.
// Feature width fixed at 128.
// ---------------------------------------------------------------------------
__global__ __launch_bounds__(256) void bias_act_kernel(
    float* __restrict__ h, const float* __restrict__ b1,
    const float* __restrict__ b2, size_t n, int do_leaky) {
  size_t i = (size_t)blockIdx.x * blockDim.x + threadIdx.x;
  if (i >= n) return;
  const int col = (int)(i & 127);
  float v = h[i] + b1[col];
  if (b2) v += b2[col];
  if (do_leaky) v = (v > 0.0f) ? v : 0.01f * v;
  h[i] = v;
}

// ---------------------------------------------------------------------------
extern "C" void kernel_launch(void* const* d_in, const int* in_sizes, int n_in,
                              void* d_out, int out_size, void* d_ws,
                              size_t ws_size, hipStream_t stream) {
  (void)n_in; (void)ws_size;
  const float* x_chem = (const float*)d_in[0];
  const float* x_gene = (const float*)d_in[1];
  const int* src_cc = (const int*)d_in[2];
  const int* dst_cc = (const int*)d_in[3];
  const int* src_cg = (const int*)d_in[4];
  const int* dst_cg = (const int*)d_in[5];
  const int* src_gc = (const int*)d_in[6];
  const int* dst_gc = (const int*)d_in[7];
  const float* W1_cc = (const float*)d_in[8];
  const float* W1_cg = (const float*)d_in[9];
  const float* W1_gc = (const float*)d_in[10];
  const float* W2_cc = (const float*)d_in[11];
  const float* W2_cg = (const float*)d_in[12];
  const float* W2_gc = (const float*)d_in[13];
  const float* b1_cc = (const float*)d_in[14];
  const float* b1_cg = (const float*)d_in[15];
  const float* b1_gc = (const float*)d_in[16];
  const float* b2_cc = (const float*)d_in[17];
  const float* b2_cg = (const float*)d_in[18];
  const float* b2_gc = (const float*)d_in[19];

  const int Nc  = in_sizes[0] / 128;
  const int Ng  = in_sizes[1] / 128;
  const int Ecc = in_sizes[2];
  const int Ecg = in_sizes[4];
  const int Egc = in_sizes[6];

  // Workspace carve-out (floats)
  float* w = (float*)d_ws;
  float* degS_cc = w; w += Nc;   // out-degree of chem over cc edges
  float* degD_cc = w; w += Nc;   // in-degree  of chem over cc edges
  float* degS_cg = w; w += Nc;   // out-degree of chem over cg edges
  float* degD_cg = w; w += Ng;   // in-degree  of gene over cg edges
  float* degS_gc = w; w += Ng;   // out-degree of gene over gc edges
  float* degD_gc = w; w += Nc;   // in-degree  of chem over gc edges
  float* tmpC = w; w += (size_t)Nc * 128;   // chem-side feature scratch
  float* tmpG = w; w += (size_t)Ng * 128;   // gene-side feature scratch
  float* aggC = w; w += (size_t)Nc * 128;   // h1_chem accumulator
  float* aggG = w; w += (size_t)Ng * 128;   // h1_gene accumulator

  // Zero all accumulators (capture-safe memsets)
  hipMemsetAsync(d_ws, 0, (size_t)(4 * Nc + 2 * Ng) * sizeof(float), stream);
  hipMemsetAsync(aggC, 0, (size_t)Nc * 128 * sizeof(float), stream);
  hipMemsetAsync(aggG, 0, (size_t)Ng * 128 * sizeof(float), stream);
  hipMemsetAsync(d_out, 0, (size_t)out_size * sizeof(float), stream);

  // Degrees (per relation, per endpoint)
  deg_count_kernel<<<(Ecc + 255) / 256, 256, 0, stream>>>(src_cc, degS_cc, Ecc);
  deg_count_kernel<<<(Ecc + 255) / 256, 256, 0, stream>>>(dst_cc, degD_cc, Ecc);
  deg_count_kernel<<<(Ecg + 255) / 256, 256, 0, stream>>>(src_cg, degS_cg, Ecg);
  deg_count_kernel<<<(Ecg + 255) / 256, 256, 0, stream>>>(dst_cg, degD_cg, Ecg);
  deg_count_kernel<<<(Egc + 255) / 256, 256, 0, stream>>>(src_gc, degS_gc, Egc);
  deg_count_kernel<<<(Egc + 255) / 256, 256, 0, stream>>>(dst_gc, degD_gc, Egc);

  const int sc_cc = (Ecc + 7) / 8;  // wave-per-edge, 8 waves per 256-thread block
  const int sc_cg = (Ecg + 7) / 8;
  const int sc_gc = (Egc + 7) / 8;

  // ---- Layer 1 ----
  // cc: chem -> chem
  gemm_norm_wmma_kernel<<<Nc / 16, 256, 0, stream>>>(x_chem, degS_cc, W1_cc, tmpC);
  scatter_edges_kernel<<<sc_cc, 256, 0, stream>>>(tmpC, src_cc, dst_cc, degD_cc, aggC, Ecc);
  // gc: gene -> chem
  gemm_norm_wmma_kernel<<<Ng / 16, 256, 0, stream>>>(x_gene, degS_gc, W1_gc, tmpG);
  scatter_edges_kernel<<<sc_gc, 256, 0, stream>>>(tmpG, src_gc, dst_gc, degD_gc, aggC, Egc);
  // cg: chem -> gene
  gemm_norm_wmma_kernel<<<Nc / 16, 256, 0, stream>>>(x_chem, degS_cg, W1_cg, tmpC);
  scatter_edges_kernel<<<sc_cg, 256, 0, stream>>>(tmpC, src_cg, dst_cg, degD_cg, aggG, Ecg);

  const size_t nC = (size_t)Nc * 128, nG = (size_t)Ng * 128;
  bias_act_kernel<<<(int)((nC + 255) / 256), 256, 0, stream>>>(aggC, b1_cc, b1_gc, nC, 1);
  bias_act_kernel<<<(int)((nG + 255) / 256), 256, 0, stream>>>(aggG, b1_cg, nullptr, nG, 1);

  // ---- Layer 2 ----
  gemm_norm_wmma_kernel<<<Nc / 16, 256, 0, stream>>>(aggC, degS_cc, W2_cc, tmpC);
  gemm_norm_wmma_kernel<<<Ng / 16, 256, 0, stream>>>(aggG, degS_gc, W2_gc, tmpG);
  float* out = (float*)d_out;
  scatter_edges_kernel<<<sc_cc, 256, 0, stream>>>(tmpC, src_cc, dst_cc, degD_cc, out, Ecc);
  scatter_edges_kernel<<<sc_gc, 256, 0, stream>>>(tmpG, src_gc, dst_gc, degD_gc, out, Egc);
  bias_act_kernel<<<(int)((nC + 255) / 256), 256, 0, stream>>>(out, b2_cc, b2_gc, nC, 0);
}